// RSLDS_50431505990277
// MI455X (gfx1250) — compile-verified
//
#include <hip/hip_runtime.h>
#include <math.h>

#define Bb 8
#define Tt 1024
#define Nn 32
#define Kk 8
#define Hh 8
#define INn 2000

typedef __attribute__((ext_vector_type(2))) float v2f;
typedef __attribute__((ext_vector_type(8))) float v8f;

// width-8 log-sum-exp across the 8-lane group containing this lane (wave32)
__device__ __forceinline__ float lse8(float v) {
  float m = v;
  m = fmaxf(m, __shfl_xor(m, 1, 8));
  m = fmaxf(m, __shfl_xor(m, 2, 8));
  m = fmaxf(m, __shfl_xor(m, 4, 8));
  float e = __expf(v - m);
  e += __shfl_xor(e, 1, 8);
  e += __shfl_xor(e, 2, 8);
  e += __shfl_xor(e, 4, 8);
  return m + __logf(e);
}

// ---------------------------------------------------------------------------
// Kernel 1: U[b,t,n] = sum_i U_raw[b,t,i]*U_scale[i]*C_syn[n,i]
// (8192 x 2000) @ (2000 x 32) via V_WMMA_F32_16X16X4_F32, one 16x16 tile/wave.
// A layout (16x4 f32): lanes 0-15 -> M=lane, VGPR0=K0,VGPR1=K1;
//                      lanes 16-31 -> M=lane-16, VGPR0=K2,VGPR1=K3.
// B layout (4x16): mirrored (rows K striped the same way, N=lane&15).
// D layout: VGPR r -> M = r + 8*(lane>=16), N = lane&15.
// ---------------------------------------------------------------------------
__global__ __launch_bounds__(32) void k_gemm_u(const float* __restrict__ U_raw,
                                               const float* __restrict__ U_scale,
                                               const float* __restrict__ C_syn,
                                               float* __restrict__ U) {
  const int mt = blockIdx.x;            // 512 row tiles over B*T
  const int nt = blockIdx.y;            // 2 col tiles over N
  const int lane = threadIdx.x;
  const int l15 = lane & 15;
  const int koff = (lane >> 4) << 1;    // 0 or 2

  const float* arow = U_raw + (size_t)(mt * 16 + l15) * INn;
  const float* brow = C_syn + (size_t)(nt * 16 + l15) * INn;

  v8f acc = {};
  for (int kk = 0; kk < INn; kk += 4) {
    const int ka = kk + koff;
    v2f a, bv;
    a.x = arow[ka]     * U_scale[ka];
    a.y = arow[ka + 1] * U_scale[ka + 1];
    bv.x = brow[ka];
    bv.y = brow[ka + 1];
    acc = __builtin_amdgcn_wmma_f32_16x16x4_f32(false, a, false, bv,
                                                (short)0, acc, false, false);
  }
  const int mbase = mt * 16 + ((lane >> 4) << 3);
  const int col = nt * 16 + l15;
#pragma unroll
  for (int r = 0; r < 8; ++r)
    U[(size_t)(mbase + r) * Nn + col] = acc[r];
}

// ---------------------------------------------------------------------------
// Kernel 2: log_B[b,t,n,k] and log_A[b,t,n,q,k]; one thread per (b,t,n,k).
// All reductions (over h, g, and the q-axis LSE) are thread-local.
// ---------------------------------------------------------------------------
__global__ __launch_bounds__(256) void k_likelihood(
    const float* __restrict__ X_inf, const float* __restrict__ U,
    const float* __restrict__ W_xx, const float* __restrict__ W_xu,
    const float* __restrict__ b_x, const float* __restrict__ W_zx,
    const float* __restrict__ W_zu, const float* __restrict__ b_z,
    const int* __restrict__ temp_p,
    float* __restrict__ logB, float* __restrict__ logA) {
  const int idx = blockIdx.x * 256 + threadIdx.x;  // over B*T*N*K = 2^21
  const int k = idx & (Kk - 1);
  const int n = (idx >> 3) & (Nn - 1);
  const int t = (idx >> 8) & (Tt - 1);
  const int b = idx >> 18;
  const float invT = 1.0f / (float)temp_p[0];

  float xold[Hh], xnew[Hh];
  const float* xn = X_inf + (size_t)((b * Tt + t) * Nn + n) * Hh;
#pragma unroll
  for (int h = 0; h < Hh; ++h) xnew[h] = xn[h];
  if (t > 0) {
    const float* xo = xn - (size_t)Nn * Hh;
#pragma unroll
    for (int h = 0; h < Hh; ++h) xold[h] = xo[h];
  } else {
#pragma unroll
    for (int h = 0; h < Hh; ++h) xold[h] = 0.f;
  }
  const float u = U[(b * Tt + t) * Nn + n];
  const int nk = n * Kk + k;

  // log_B = sum_h (W_xx[n,k,h,:]·xold + W_xu[n,k,h]*u + b_x[n,k,h] - xnew[h])^2
  const float* wxx = W_xx + (size_t)nk * Hh * Hh;
  const float* wxu = W_xu + (size_t)nk * Hh;
  const float* bx  = b_x  + (size_t)nk * Hh;
  float sq = 0.f;
#pragma unroll
  for (int h = 0; h < Hh; ++h) {
    float e = bx[h] + wxu[h] * u;
#pragma unroll
    for (int g = 0; g < Hh; ++g) e += wxx[h * Hh + g] * xold[g];
    const float d = e - xnew[h];
    sq += d * d;
  }
  logB[idx] = sq;

  // log_A[b,t,n,q,k]; t==0 is uniform; otherwise X_part = X_inf[b,t-1] = xold
  float* la = logA + (size_t)((b * Tt + t) * Nn + n) * Kk * Kk + k;
  if (t == 0) {
#pragma unroll
    for (int q = 0; q < Kk; ++q) la[q * Kk] = -2.0794415416798357f;  // log(1/8)
  } else {
    const float* wzx = W_zx + (size_t)nk * Kk * Hh;
    const float* wzu = W_zu + (size_t)nk * Kk;
    const float* bz  = b_z  + (size_t)nk * Kk;
    float a[Kk];
    float m = -3.0e38f;
#pragma unroll
    for (int q = 0; q < Kk; ++q) {
      float s = bz[q] + wzu[q] * u;
#pragma unroll
      for (int h = 0; h < Hh; ++h) s += wzx[q * Hh + h] * xold[h];
      s *= invT;
      a[q] = s;
      m = fmaxf(m, s);
    }
    float se = 0.f;
#pragma unroll
    for (int q = 0; q < Kk; ++q) se += __expf(a[q] - m);
    const float lse = m + __logf(se);
#pragma unroll
    for (int q = 0; q < Kk; ++q) la[q * Kk] = a[q] - lse;
  }
}

// ---------------------------------------------------------------------------
// Kernel 3: forward + backward scans. One chain (b,n) per 8 lanes, state=lane%8.
// logA/logB are L2-resident (67+8 MB << 192 MB L2) so per-step loads are cheap.
// ---------------------------------------------------------------------------
__global__ __launch_bounds__(256) void k_scan(
    const float* __restrict__ lpz1,
    const float* __restrict__ logB, const float* __restrict__ logA,
    float* __restrict__ out) {
  const int tid = blockIdx.x * 256 + threadIdx.x;  // B*N*K = 2048 threads
  const int chain = tid >> 3;                      // b*N + n
  const int s = tid & 7;                           // state index
  const int b = chain / Nn;
  const int n = chain - b * Nn;
  const size_t cK = (size_t)(b * Tt * Nn + n) * Kk;   // ((b*T+0)*N+n)*K
  const float* lB = logB + cK;
  const float* lA = logA + cK * Kk;
  const int stB = Nn * Kk;        // 256
  const int stA = Nn * Kk * Kk;   // 2048
  const size_t S = (size_t)Bb * Tt * Nn * Kk;
  float* fp = out + cK;
  float* bp = out + S + cK;

  // ---- forward: thread s plays role q ----
  float p = lpz1[s] + lB[s];
  p -= lse8(p);
  fp[s] = p;
  for (int t = 1; t < Tt; ++t) {
    const float* At = lA + (size_t)t * stA + s * Kk;   // row q=s, contiguous in k
    const float bq = lB[t * stB + s];
    float x[Kk];
    float m = -3.0e38f;
#pragma unroll
    for (int k = 0; k < Kk; ++k) {
      const float v = At[k] + __shfl(p, k, 8);         // prev[k] from lane k
      x[k] = v;
      m = fmaxf(m, v);
    }
    float se = 0.f;
#pragma unroll
    for (int k = 0; k < Kk; ++k) se += __expf(x[k] - m);
    const float uq = bq + m + __logf(se);
    p = uq - lse8(uq);
    fp[t * stB + s] = p;
  }

  // ---- backward: thread s plays role k ----
  float c = 0.f;
  bp[(Tt - 1) * stB + s] = 0.f;
  for (int t = Tt - 1; t >= 1; --t) {
    const float* At = lA + (size_t)t * stA + s;        // column k=s, stride K over q
    const float* Bt = lB + t * stB;
    float x[Kk];
    float m = -3.0e38f;
#pragma unroll
    for (int q = 0; q < Kk; ++q) {
      const float v = __shfl(c, q, 8) + At[q * Kk] + Bt[q];
      x[q] = v;
      m = fmaxf(m, v);
    }
    float se = 0.f;
#pragma unroll
    for (int q = 0; q < Kk; ++q) se += __expf(x[q] - m);
    const float u = m + __logf(se);
    c = u - lse8(u);
    bp[(t - 1) * stB + s] = c;
  }
}

// ---------------------------------------------------------------------------
// Kernel 4: gamma1 + gamma2. One wave per (b,t,n): 64 (q,k) cells = 2/lane.
// ---------------------------------------------------------------------------
__global__ __launch_bounds__(256) void k_gamma(
    const float* __restrict__ logB, const float* __restrict__ logA,
    float* __restrict__ out) {
  const size_t S = (size_t)Bb * Tt * Nn * Kk;
  const int w = blockIdx.x * 8 + (threadIdx.x >> 5);  // w = (b*T+t)*N+n
  const int lane = threadIdx.x & 31;
  const int t = (w >> 5) & (Tt - 1);
  const size_t baseK = (size_t)w * Kk;
  const size_t base64 = (size_t)w * (Kk * Kk);

  const float* fp = out;
  const float* bp = out + S;
  float* g1 = out + 2 * S;
  float* g2 = out + 3 * S;

  // gamma1 on lanes 0-7 (width-8 LSE keeps groups independent)
  const float g = fp[baseK + (lane & 7)] + bp[baseK + (lane & 7)];
  const float l1 = lse8(g);
  if (lane < 8) g1[baseK + lane] = g - l1;

  if (t == 0) {
    g2[base64 + lane] = 0.f;
    g2[base64 + lane + 32] = 0.f;
    return;
  }
  // element e = q*8+k ; this lane owns e=lane (q=lane>>3) and e=lane+32 (q+4)
  const int q0 = lane >> 3;
  const int k0 = lane & 7;
  const float* At = logA + base64;
  const float fprev = fp[baseK - (size_t)(Nn * Kk) + k0];   // forward[t-1]
  float v0 = fprev + bp[baseK + q0]     + At[q0 * Kk + k0]       + logB[baseK + q0];
  float v1 = fprev + bp[baseK + q0 + 4] + At[(q0 + 4) * Kk + k0] + logB[baseK + q0 + 4];

  float m = fmaxf(v0, v1);
#pragma unroll
  for (int d = 1; d < 32; d <<= 1) m = fmaxf(m, __shfl_xor(m, d, 32));
  float e = __expf(v0 - m) + __expf(v1 - m);
#pragma unroll
  for (int d = 1; d < 32; d <<= 1) e += __shfl_xor(e, d, 32);
  const float lse = m + __logf(e);
  g2[base64 + lane] = v0 - lse;
  g2[base64 + lane + 32] = v1 - lse;
}

// ---------------------------------------------------------------------------
extern "C" void kernel_launch(void* const* d_in, const int* in_sizes, int n_in,
                              void* d_out, int out_size, void* d_ws, size_t ws_size,
                              hipStream_t stream) {
  const float* U_raw   = (const float*)d_in[0];
  const float* lpz1    = (const float*)d_in[1];
  const float* C_syn   = (const float*)d_in[2];
  const float* U_scale = (const float*)d_in[3];
  const float* W_zu    = (const float*)d_in[4];
  const float* W_zx    = (const float*)d_in[5];
  const float* W_xx    = (const float*)d_in[6];
  const float* W_xu    = (const float*)d_in[7];
  const float* b_z     = (const float*)d_in[8];
  const float* b_x     = (const float*)d_in[9];
  const float* X_inf   = (const float*)d_in[10];
  const int*   temp    = (const int*)d_in[11];

  float* out = (float*)d_out;
  float* ws  = (float*)d_ws;
  float* U    = ws;                                     // B*T*N        (1 MB)
  float* logB = U + (size_t)Bb * Tt * Nn;               // B*T*N*K    (8.4 MB)
  float* logA = logB + (size_t)Bb * Tt * Nn * Kk;       // B*T*N*K*K   (67 MB)

  dim3 gGemm((Bb * Tt) / 16, Nn / 16);
  k_gemm_u<<<gGemm, 32, 0, stream>>>(U_raw, U_scale, C_syn, U);

  k_likelihood<<<(Bb * Tt * Nn * Kk) / 256, 256, 0, stream>>>(
      X_inf, U, W_xx, W_xu, b_x, W_zx, W_zu, b_z, temp, logB, logA);

  k_scan<<<(Bb * Nn * Kk) / 256, 256, 0, stream>>>(lpz1, logB, logA, out);

  k_gamma<<<(Bb * Tt * Nn) / 8, 256, 0, stream>>>(logB, logA, out);
}